// Threshold_weights7_52699248721953
// MI455X (gfx1250) — compile-verified
//
#include <hip/hip_runtime.h>

#define C_DIM 128
#define NPRED 8
#define GROUP 32                         // rows per wave (one row per lane)
#define WPB 4                            // waves per block
#define THREADS (WPB * 32)
#define ROW_BYTES (C_DIM * 4)            // 512
#define ROW_PAD 16                       // bank-conflict padding per row
#define ROW_STRIDE (ROW_BYTES + ROW_PAD) // 528 (multiple of 16)
#define BUF_BYTES (GROUP * ROW_STRIDE)   // 16896 per buffer

typedef unsigned long long u64;
typedef unsigned int u32;

__global__ __launch_bounds__(THREADS)
void margin_softmax_kernel(const float* __restrict__ p0, const float* __restrict__ p1,
                           const float* __restrict__ p2, const float* __restrict__ p3,
                           const float* __restrict__ p4, const float* __restrict__ p5,
                           const float* __restrict__ p6, const float* __restrict__ p7,
                           const int* __restrict__ targets,
                           float* __restrict__ out_thr,   // d_out + 1, [N][8]
                           float* __restrict__ block_max, // d_ws, one per block
                           int n) {
    // per wave: 2 ping-pong buffers of 32 padded rows
    __shared__ __align__(16) char lbuf[2 * WPB * BUF_BYTES]; // 135168 B
    __shared__ float s_wmax[WPB];

    const int lane  = threadIdx.x & 31;
    const int wave  = threadIdx.x >> 5;
    const int gwave = blockIdx.x * WPB + wave;
    const int gbase = gwave * GROUP;     // first row of this wave's group

    const float* const preds[NPRED] = {p0, p1, p2, p3, p4, p5, p6, p7};

    char* wbuf = &lbuf[wave * 2 * BUF_BYTES];
    const u32 lds0 = (u32)(u64)(uintptr_t)wbuf;

    float wmax = -__builtin_inff();

    int rows = n - gbase;
    if (rows > GROUP) rows = GROUP;

    if (rows == GROUP) {
        const u64 gro = (u64)gbase * ROW_BYTES;

        // stage predictor p's 32 rows into buffer b:
        // instruction k copies row k (512 B contiguous, 16 B per lane)
        auto issue = [&](int p, int b) {
            const char* base = (const char*)preds[p] + gro + (u64)(lane * 16);
            u32 l = lds0 + (u32)b * BUF_BYTES + (u32)(lane * 16);
#pragma unroll
            for (int k = 0; k < GROUP; ++k) {
                asm volatile("global_load_async_to_lds_b128 %0, %1, off"
                             :: "v"(l), "v"((u64)(uintptr_t)(base + (u64)k * ROW_BYTES))
                             : "memory");
                l += ROW_STRIDE;
            }
        };

        issue(0, 0);
        const int t = targets[gbase + lane];   // coalesced, per-lane row target

        float mar[NPRED];
#pragma unroll
        for (int p = 0; p < NPRED; ++p) {
            if (p + 1 < NPRED) {
                issue(p + 1, (p + 1) & 1);
                asm volatile("s_wait_asynccnt 32" ::: "memory");
            } else {
                asm volatile("s_wait_asynccnt 0" ::: "memory");
            }

            // lane scans its own padded row: 32 x ds_load_b128, running top-2
            const char* myrow = wbuf + (p & 1) * BUF_BYTES + lane * ROW_STRIDE;
            float m1 = -__builtin_inff();
            float m2 = -__builtin_inff();
#pragma unroll
            for (int k = 0; k < GROUP; ++k) {
                float4 v = *(const float4*)(myrow + k * 16);
                float a1 = fmaxf(v.x, v.y), a2 = fminf(v.x, v.y);
                float b1 = fmaxf(v.z, v.w), b2 = fminf(v.z, v.w);
                float c1 = fmaxf(a1, b1);
                float c2 = fmaxf(fminf(a1, b1), fmaxf(a2, b2));
                float hi = fmaxf(m1, c1);
                m2 = fmaxf(fminf(m1, c1), fmaxf(m2, c2));
                m1 = hi;
            }
            float tv = *(const float*)(myrow + t * 4);  // target logit, per-lane
            mar[p] = (tv == m1) ? (m1 - m2) : 0.0f;
            if (p < 7) wmax = fmaxf(wmax, m1);          // global max: outputs1..7 only
        }

        // softmax over 8 margins, T = 2 (fully lane-local)
        float mx = mar[0];
#pragma unroll
        for (int p = 1; p < NPRED; ++p) mx = fmaxf(mx, mar[p]);
        float e[NPRED];
        float s = 0.0f;
#pragma unroll
        for (int p = 0; p < NPRED; ++p) {
            e[p] = __expf((mar[p] - mx) * 0.5f);
            s += e[p];
        }
        const float inv = 1.0f / s;

        const u64 ob = (u64)(gbase + lane) * NPRED;  // out_thr is only 4B aligned
#pragma unroll
        for (int j = 0; j < NPRED; ++j)
            out_thr[ob + j] = e[j] * inv;
    } else if (rows > 0 && lane < rows) {
        // rare tail group: direct global scan, one row per active lane
        const int row = gbase + lane;
        const int t = targets[row];
        float mar[NPRED];
#pragma unroll
        for (int p = 0; p < NPRED; ++p) {
            const float* r = preds[p] + (u64)row * C_DIM;
            float m1 = -__builtin_inff();
            float m2 = -__builtin_inff();
            for (int c = 0; c < C_DIM; ++c) {
                float x = r[c];
                float hi = fmaxf(m1, x);
                m2 = fmaxf(fminf(m1, x), m2);
                m1 = hi;
            }
            float tv = r[t];
            mar[p] = (tv == m1) ? (m1 - m2) : 0.0f;
            if (p < 7) wmax = fmaxf(wmax, m1);
        }
        float mx = mar[0];
        for (int p = 1; p < NPRED; ++p) mx = fmaxf(mx, mar[p]);
        float e[NPRED];
        float s = 0.0f;
        for (int p = 0; p < NPRED; ++p) { e[p] = __expf((mar[p] - mx) * 0.5f); s += e[p]; }
        const float inv = 1.0f / s;
        const u64 ob = (u64)row * NPRED;
        for (int j = 0; j < NPRED; ++j) out_thr[ob + j] = e[j] * inv;
    }

    // scalar output: reduce per-lane maxima -> wave -> block -> d_ws
#pragma unroll
    for (int off = 16; off >= 1; off >>= 1)
        wmax = fmaxf(wmax, __shfl_xor(wmax, off, 32));
    if (lane == 0) s_wmax[wave] = wmax;
    __syncthreads();
    if (threadIdx.x == 0) {
        float bm = s_wmax[0];
#pragma unroll
        for (int w = 1; w < WPB; ++w) bm = fmaxf(bm, s_wmax[w]);
        block_max[blockIdx.x] = bm;
    }
}

__global__ __launch_bounds__(1024)
void reduce_max_kernel(const float* __restrict__ part, int nb,
                       float* __restrict__ out0) {
    float m = -__builtin_inff();
    for (int i = threadIdx.x; i < nb; i += blockDim.x)
        m = fmaxf(m, part[i]);
#pragma unroll
    for (int off = 16; off >= 1; off >>= 1)
        m = fmaxf(m, __shfl_xor(m, off, 32));

    __shared__ float sm[32];
    const int lane = threadIdx.x & 31;
    const int wv   = threadIdx.x >> 5;
    if (lane == 0) sm[wv] = m;
    __syncthreads();
    if (threadIdx.x == 0) {
        const int nw = blockDim.x >> 5;
        float r = sm[0];
        for (int w = 1; w < nw; ++w) r = fmaxf(r, sm[w]);
        out0[0] = r;
    }
}

extern "C" void kernel_launch(void* const* d_in, const int* in_sizes, int n_in,
                              void* d_out, int out_size, void* d_ws, size_t ws_size,
                              hipStream_t stream) {
    const float* p[8];
    for (int i = 0; i < 8; ++i) p[i] = (const float*)d_in[i];
    const int* targets = (const int*)d_in[8];
    const int n = in_sizes[8];

    float* out  = (float*)d_out;   // out[0] = max_preds; out[1..] = [N][8] softmax
    float* bmax = (float*)d_ws;    // per-block partial maxima

    const int ngroups = (n + GROUP - 1) / GROUP;
    const int nblocks = (ngroups + WPB - 1) / WPB;

    margin_softmax_kernel<<<nblocks, THREADS, 0, stream>>>(
        p[0], p[1], p[2], p[3], p[4], p[5], p[6], p[7],
        targets, out + 1, bmax, n);

    reduce_max_kernel<<<1, 1024, 0, stream>>>(bmax, nblocks, out);
}